// YoloLoss_29394756174227
// MI455X (gfx1250) — compile-verified
//
#include <hip/hip_runtime.h>

// YOLO loss, fp32, memory-bound streaming reduction for gfx1250 (wave32).
// Roofline: 2 x 104.9 MB read once at 23.3 TB/s -> ~9 us floor; compute is
// ~0.14 GFLOP (negligible). Strategy: coalesced non-temporal b128 loads,
// per-wave reduction via V_WMMA_F32_16X16X4_F32, deterministic 2-pass sum.

typedef __attribute__((ext_vector_type(2))) float v2f;
typedef __attribute__((ext_vector_type(4))) float v4f;
typedef __attribute__((ext_vector_type(8))) float v8f;

#define COORD 5.0f
#define NOOBJ 0.5f
#define INV_BLOCKS (1.0f / 64.0f)

// Full 32-lane sum using one f32 WMMA (wave32, EXEC must be all ones).
// A(16x4): lane L<16 -> A[L][0]=v, A[L][1]=0 ; lane L>=16 -> A[L-16][2]=v, [3]=0
// B(4x16) = all ones  =>  D[m][n] = v[m] + v[m+16]  (independent of n)
// Sum of the 8 D vgprs: lanes 0-15 hold rows 0-7, lanes 16-31 hold rows 8-15.
// One xor-16 shuffle completes the reduction; all lanes end with the total.
__device__ __forceinline__ float wave_sum32(float v) {
    v2f a; a[0] = v;    a[1] = 0.0f;
    v2f b; b[0] = 1.0f; b[1] = 1.0f;
    v8f c = {};
    v8f d = __builtin_amdgcn_wmma_f32_16x16x4_f32(
        /*neg_a=*/false, a, /*neg_b=*/false, b,
        /*c_mod=*/(short)0, c, /*reuse_a=*/false, /*reuse_b=*/false);
    float s = d[0] + d[1] + d[2] + d[3] + d[4] + d[5] + d[6] + d[7];
    s += __shfl_xor(s, 16, 32);
    return s;
}

__global__ __launch_bounds__(256) void yolo_pass1(const float* __restrict__ inp,
                                                  const float* __restrict__ tgt,
                                                  float* __restrict__ partial) {
    // Cells: N = 256*64*64 = 1,048,576; each thread handles a "quad" of 4
    // consecutive cells inside one channel plane (4096 cells/plane).
    const int quad  = blockIdx.x * 256 + threadIdx.x;   // [0, 262144)
    const int batch = quad >> 10;                       // plane has 1024 quads
    const int r4    = (quad & 1023) << 2;               // cell offset in plane
    const long base = (long)batch * (25L * 4096L) + r4;

    const v4f* ip = (const v4f*)(inp + base);           // channel c at c*1024
    const v4f* tp = (const v4f*)(tgt + base);

    v4f p0 = __builtin_nontemporal_load(&ip[0 * 1024]);
    v4f p1 = __builtin_nontemporal_load(&ip[1 * 1024]);
    v4f p2 = __builtin_nontemporal_load(&ip[2 * 1024]);
    v4f p3 = __builtin_nontemporal_load(&ip[3 * 1024]);
    v4f p4 = __builtin_nontemporal_load(&ip[4 * 1024]);
    v4f t0 = __builtin_nontemporal_load(&tp[0 * 1024]);
    v4f t1 = __builtin_nontemporal_load(&tp[1 * 1024]);
    v4f t2 = __builtin_nontemporal_load(&tp[2 * 1024]);
    v4f t3 = __builtin_nontemporal_load(&tp[3 * 1024]);
    v4f t4 = __builtin_nontemporal_load(&tp[4 * 1024]);

    float acc = 0.0f;
    float mm[4];
#pragma unroll
    for (int i = 0; i < 4; ++i) {
        const float P0 = p0[i], P1 = p1[i], P2 = p2[i], P3 = p3[i], P4 = p4[i];
        const float T0 = t0[i], T1 = t1[i], T2 = t2[i], T3 = t3[i], T4 = t4[i];
        const float m = (T4 > 0.0f) ? 1.0f : 0.0f;
        mm[i] = m;

        // IoU (forward value; stop_gradient irrelevant)
        const float x1a = P0 * INV_BLOCKS - P2 * 0.5f;
        const float x2a = P0 * INV_BLOCKS + P2 * 0.5f;
        const float y1a = P1 * INV_BLOCKS - P3 * 0.5f;
        const float y2a = P1 * INV_BLOCKS + P3 * 0.5f;
        const float x1b = T0 * INV_BLOCKS - T2 * 0.5f;
        const float x2b = T0 * INV_BLOCKS + T2 * 0.5f;
        const float y1b = T1 * INV_BLOCKS - T3 * 0.5f;
        const float y2b = T1 * INV_BLOCKS + T3 * 0.5f;
        const float dx = fminf(x2a, x2b) - fmaxf(x1a, x1b);
        const float dy = fminf(y2a, y2b) - fmaxf(y1a, y1b);
        const float inter = dx * dy;
        const float uni   = P2 * P3 + T2 * T3 - inter;
        const float iou   = (dx > 0.0f && dy > 0.0f) ? (inter / uni) : 0.0f;

        const float d0 = P0 - T0, d1 = P1 - T1;
        acc += COORD * m * (d0 * d0 + d1 * d1);
        const float ds2 = sqrtf(P2) - sqrtf(T2);
        const float ds3 = sqrtf(P3) - sqrtf(T3);
        acc += COORD * m * (ds2 * ds2 + ds3 * ds3);
        const float dconf = P4 - iou;
        acc += m * dconf * dconf + NOOBJ * (1.0f - m) * P4 * P4;
    }

    // Class channels 5..24
#pragma unroll
    for (int ch = 5; ch < 25; ++ch) {
        v4f pc = __builtin_nontemporal_load(&ip[ch * 1024]);
        v4f tc = __builtin_nontemporal_load(&tp[ch * 1024]);
#pragma unroll
        for (int i = 0; i < 4; ++i) {
            const float d = pc[i] - tc[i];
            acc += mm[i] * d * d;
        }
    }

    // Block reduction: WMMA per wave, then 8 wave sums via LDS.
    const float wsum = wave_sum32(acc);            // all lanes hold wave total
    __shared__ float lds[8];
    const int lane = threadIdx.x & 31;
    const int wid  = threadIdx.x >> 5;
    if (lane == 0) lds[wid] = wsum;
    __syncthreads();
    if (threadIdx.x == 0) {
        float s = 0.0f;
#pragma unroll
        for (int i = 0; i < 8; ++i) s += lds[i];
        partial[blockIdx.x] = s;
    }
}

__global__ __launch_bounds__(256) void yolo_pass2(const float* __restrict__ partial,
                                                  float* __restrict__ out) {
    // 1024 partials, fixed-order deterministic reduction.
    float v = partial[threadIdx.x] + partial[threadIdx.x + 256] +
              partial[threadIdx.x + 512] + partial[threadIdx.x + 768];
    const float wsum = wave_sum32(v);
    __shared__ float lds[8];
    const int lane = threadIdx.x & 31;
    const int wid  = threadIdx.x >> 5;
    if (lane == 0) lds[wid] = wsum;
    __syncthreads();
    if (threadIdx.x == 0) {
        float s = 0.0f;
#pragma unroll
        for (int i = 0; i < 8; ++i) s += lds[i];
        out[0] = s;
    }
}

extern "C" void kernel_launch(void* const* d_in, const int* in_sizes, int n_in,
                              void* d_out, int out_size, void* d_ws, size_t ws_size,
                              hipStream_t stream) {
    const float* inp = (const float*)d_in[0];   // 'input'  (256,25,64,64) f32
    const float* tgt = (const float*)d_in[1];   // 'target' (256,25,64,64) f32
    float* out     = (float*)d_out;             // scalar f32
    float* partial = (float*)d_ws;              // 1024 floats = 4 KB scratch

    yolo_pass1<<<1024, 256, 0, stream>>>(inp, tgt, partial);
    yolo_pass2<<<1,    256, 0, stream>>>(partial, out);
}